// WaveletBlock1D_15247133901360
// MI455X (gfx1250) — compile-verified
//
#include <hip/hip_runtime.h>
#include <math.h>

typedef __attribute__((ext_vector_type(2))) float v2f;
typedef __attribute__((ext_vector_type(8))) float v8f;

#define B_   32
#define L_   16384
#define C_   64
#define LC_  2048            // L / 2^3
#define KISQ8 0.3535533905932738f   // (1/sqrt(2))^3

#define XS_STRIDE 68         // padded row stride (floats): 272B, 16B aligned, bank-conflict free
#define A_STRIDE  68

// LDS layout (floats)
#define XS_FLOATS   (256 * XS_STRIDE)          // x tile: rows (b*8+j), cols c
#define AARR_FLOATS (32 * A_STRIDE)            // a3 / d3 / cA / cD
#define W_FLOATS    (64 * 64)                  // Wa / Wd / Wskip
#define SMEM_FLOATS (XS_FLOATS + 4 * AARR_FLOATS + 3 * W_FLOATS + 64)

__device__ __forceinline__ float gelu_tanh(float x) {
    // jax.nn.gelu (approximate=True)
    float x3 = x * x * x;
    float t  = tanhf(0.7978845608028654f * (x + 0.044715f * x3));
    return 0.5f * x * (1.0f + t);
}

__global__ void __launch_bounds__(256, 2)
wavelet_block1d_kernel(const float* __restrict__ x,
                       const float* __restrict__ w_approx,
                       const float* __restrict__ w_detail,
                       const float* __restrict__ W_skip,
                       const float* __restrict__ b_skip,
                       float* __restrict__ out)
{
    extern __shared__ float smem[];
    float* Xs  = smem;                         // [256][68]
    float* a3s = Xs  + XS_FLOATS;              // [32][68]
    float* d3s = a3s + AARR_FLOATS;
    float* cAs = d3s + AARR_FLOATS;            // k*(A - a3)
    float* cDs = cAs + AARR_FLOATS;            // k*(D - d3)
    float* Was = cDs + AARR_FLOATS;            // [64][64]  (k=i rows, n=o cols)
    float* Wds = Was + W_FLOATS;
    float* Wss = Wds + W_FLOATS;
    float* bss = Wss + W_FLOATS;               // [64]

    const int l  = blockIdx.x;                 // 0..2047
    const int t  = threadIdx.x;                // 0..255
    const int wv   = t >> 5;                   // wave id 0..7
    const int lane = t & 31;
    const int hi   = lane >> 4;                // 0 or 1
    const int ln   = lane & 15;

    // ---------------- Phase 0: stage inputs into LDS ----------------
    // x tile: 32 b x 8 pos x 64 ch = 4096 float4, coalesced (2KB contiguous per b)
    #pragma unroll
    for (int i = 0; i < 16; ++i) {
        int idx = t + 256 * i;                 // 0..4095
        int b   = idx >> 7;
        int rem = idx & 127;
        int j   = rem >> 4;
        int c4  = rem & 15;
        int row = b * 8 + j;
        const float4* src = (const float4*)(x + (size_t)b * L_ * C_ + (size_t)(l * 8 + j) * C_);
        *(float4*)&Xs[row * XS_STRIDE + c4 * 4] = src[c4];
    }
    // per-l weight matrices: element (i,o) at flat (i*64+o)*2048 + l  (4B gather, L2-friendly across l)
    #pragma unroll
    for (int i = 0; i < 16; ++i) {
        int idx = t + 256 * i;                 // 0..4095 = i*64 + o
        Was[idx] = w_approx[(size_t)idx * LC_ + l];
        Wds[idx] = w_detail[(size_t)idx * LC_ + l];
    }
    // W_skip (row-major [c][o]) and bias
    {
        const float4* wg = (const float4*)W_skip;
        float4* ws = (float4*)Wss;
        #pragma unroll
        for (int i = 0; i < 4; ++i) ws[t + 256 * i] = wg[t + 256 * i];
    }
    if (t < 64) bss[t] = b_skip[t];
    __syncthreads();

    // ---------------- Phase 1: level-3 Haar coefficients ----------------
    #pragma unroll
    for (int i = 0; i < 8; ++i) {
        int idx = t + 256 * i;                 // 0..2047: b = idx/64, c = idx%64
        int b = idx >> 6, c = idx & 63;
        int base = (b * 8) * XS_STRIDE + c;
        float sf = Xs[base]                 + Xs[base + XS_STRIDE]
                 + Xs[base + 2 * XS_STRIDE] + Xs[base + 3 * XS_STRIDE];
        float sb = Xs[base + 4 * XS_STRIDE] + Xs[base + 5 * XS_STRIDE]
                 + Xs[base + 6 * XS_STRIDE] + Xs[base + 7 * XS_STRIDE];
        a3s[b * A_STRIDE + c] = (sf + sb) * KISQ8;
        d3s[b * A_STRIDE + c] = (sf - sb) * KISQ8;
    }
    __syncthreads();

    // ---------------- Phase A: wavelet GEMMs (M=32,K=64,N=64) via WMMA f32 ----------------
    {
        const float* As = (wv < 4) ? a3s : d3s;
        const float* Bs = (wv < 4) ? Was : Wds;
        float*       Cs = (wv < 4) ? cAs : cDs;
        int wg  = wv & 3;
        int m0  = (wg & 1) * 16;
        int n0a = (wg >> 1) * 32;              // this wave: n-tiles n0a and n0a+16

        v8f acc0 = {}; v8f acc1 = {};
        for (int k0 = 0; k0 < 64; k0 += 4) {
            v2f av = *(const v2f*)&As[(m0 + ln) * A_STRIDE + k0 + 2 * hi];
            int krow = (k0 + 2 * hi) * 64;
            v2f b0; b0[0] = Bs[krow + n0a + ln];      b0[1] = Bs[krow + 64 + n0a + ln];
            v2f b1; b1[0] = Bs[krow + n0a + 16 + ln]; b1[1] = Bs[krow + 64 + n0a + 16 + ln];
            acc0 = __builtin_amdgcn_wmma_f32_16x16x4_f32(false, av, false, b0, (short)0, acc0, false, false);
            acc1 = __builtin_amdgcn_wmma_f32_16x16x4_f32(false, av, false, b1, (short)0, acc1, false, false);
        }
        #pragma unroll
        for (int r = 0; r < 8; ++r) {
            int m  = m0 + r + 8 * hi;
            int nA = n0a + ln;
            int nB = n0a + 16 + ln;
            Cs[m * A_STRIDE + nA] = (acc0[r] - As[m * A_STRIDE + nA]) * KISQ8;
            Cs[m * A_STRIDE + nB] = (acc1[r] - As[m * A_STRIDE + nB]) * KISQ8;
        }
    }
    __syncthreads();

    // ---------------- Phase B: skip GEMM (M=256,K=64,N=64) + fused epilogue ----------------
    // wave wv owns rows [32*wv, 32*wv+32): two M-tiles, 4 N-tiles each (A-fragment reused 4x)
    for (int mi = 0; mi < 2; ++mi) {
        int m0 = (2 * wv + mi) * 16;
        v8f acc[4] = {{}, {}, {}, {}};
        for (int k0 = 0; k0 < 64; k0 += 4) {
            v2f av = *(const v2f*)&Xs[(m0 + ln) * XS_STRIDE + k0 + 2 * hi];
            int krow = (k0 + 2 * hi) * 64;
            #pragma unroll
            for (int nt = 0; nt < 4; ++nt) {
                v2f bv;
                bv[0] = Wss[krow + nt * 16 + ln];
                bv[1] = Wss[krow + 64 + nt * 16 + ln];
                acc[nt] = __builtin_amdgcn_wmma_f32_16x16x4_f32(false, av, false, bv, (short)0, acc[nt], false, false);
            }
        }
        // epilogue: y = x + corr, out = gelu(y + skip + bias); overwrite wave-private Xs rows
        #pragma unroll
        for (int nt = 0; nt < 4; ++nt) {
            #pragma unroll
            for (int r = 0; r < 8; ++r) {
                int m = m0 + r + 8 * hi;
                int b = m >> 3, j = m & 7;
                int n = nt * 16 + ln;
                float sgn = (j < 4) ? 1.0f : -1.0f;
                float val = acc[nt][r]
                          + Xs[m * XS_STRIDE + n]
                          + bss[n]
                          + cAs[b * A_STRIDE + n]
                          + sgn * cDs[b * A_STRIDE + n];
                Xs[m * XS_STRIDE + n] = gelu_tanh(val);
            }
        }
    }
    __syncthreads();

    // ---------------- Phase C: coalesced float4 stores ----------------
    #pragma unroll
    for (int i = 0; i < 16; ++i) {
        int idx = t + 256 * i;
        int b   = idx >> 7;
        int rem = idx & 127;
        int j   = rem >> 4;
        int c4  = rem & 15;
        int row = b * 8 + j;
        float4 v = *(const float4*)&Xs[row * XS_STRIDE + c4 * 4];
        *(float4*)(out + (size_t)b * L_ * C_ + (size_t)(l * 8 + j) * C_ + c4 * 4) = v;
    }
}

extern "C" void kernel_launch(void* const* d_in, const int* in_sizes, int n_in,
                              void* d_out, int out_size, void* d_ws, size_t ws_size,
                              hipStream_t stream) {
    const float* x        = (const float*)d_in[0];
    const float* w_approx = (const float*)d_in[1];
    const float* w_detail = (const float*)d_in[2];
    const float* W_skip   = (const float*)d_in[3];
    const float* b_skip   = (const float*)d_in[4];
    float* out = (float*)d_out;

    const size_t smem_bytes = (size_t)SMEM_FLOATS * sizeof(float);
    // Allow large dynamic LDS (gfx1250 WGP has 320KB); deterministic, capture-safe host call.
    hipFuncSetAttribute((const void*)wavelet_block1d_kernel,
                        hipFuncAttributeMaxDynamicSharedMemorySize, (int)smem_bytes);

    wavelet_block1d_kernel<<<dim3(LC_), dim3(256), smem_bytes, stream>>>(
        x, w_approx, w_detail, W_skip, b_skip, out);
}